// EmbeddingLayer_17892833755532
// MI455X (gfx1250) — compile-verified
//
#include <hip/hip_runtime.h>

#ifndef __has_builtin
#define __has_builtin(x) 0
#endif

#if __has_builtin(__builtin_amdgcn_global_load_async_to_lds_b128)
#define HAVE_ASYNC_LDS 1
#else
#define HAVE_ASYNC_LDS 0
#endif

// Problem geometry (from reference: B=16384, La=50, Lb=20, V=100000, D=64)
constexpr int kD    = 64;
constexpr int kLa   = 50;
constexpr int kLb   = 20;
constexpr int kOutW = 3 * kD;   // 192 floats per output row

#if HAVE_ASYNC_LDS
constexpr int kWaves = 2;       // LDS: (50+20)*256B per wave = 17.9KB -> 35.8KB/block
#else
constexpr int kWaves = 4;
#endif
constexpr int kBlock = kWaves * 32;

typedef float f32x2 __attribute__((ext_vector_type(2)));
typedef int   v4i   __attribute__((vector_size(16)));  // matches builtin param type

// Wait until at most N async global->LDS DMAs remain in flight. Async loads
// retire in order, so ASYNCcnt <= N proves all but the last N ops landed.
template <int N>
__device__ __forceinline__ void wait_async() {
#if __has_builtin(__builtin_amdgcn_s_wait_asynccnt)
    __builtin_amdgcn_s_wait_asynccnt(N);
    asm volatile("" ::: "memory");
#else
    asm volatile("s_wait_asynccnt %0" ::"i"(N) : "memory");
#endif
}

#if HAVE_ASYNC_LDS
// Stage L embedding rows (each 64 f32 = 256B) into LDS with async gather DMA.
// One global_load_async_to_lds_b128 moves 2 rows: lanes 0-15 -> row l,
// lanes 16-31 -> row l+1, each lane carrying a 16B chunk.
// Indices are wave-uniform, so load them on the scalar pipe (s_load, KMcnt)
// and pick id[l] vs id[l+1] with a v_cndmask on the lane's `half` bit —
// keeps the VMEM pipe free for the async gather DMAs themselves.
template <int L>
__device__ __forceinline__ void stage_rows_async(const int* __restrict__ ids,
                                                 const float* __restrict__ table,
                                                 float* __restrict__ lbuf,
                                                 int lane) {
    const int  sub  = lane & 15;        // 16B chunk within a row
    const bool hi   = (lane >> 4) != 0; // which of the 2 rows this lane serves
    const int  boff = sub * 4;          // float offset of this lane's chunk
#pragma unroll
    for (int l = 0; l < L; l += 2) {
        const int id0 = ids[l];          // uniform -> s_load (batched wide)
        const int id1 = ids[l + 1];      // uniform -> s_load
        const int id  = hi ? id1 : id0;  // v_cndmask
        const unsigned goff = (unsigned)id * (unsigned)kD + (unsigned)boff;
        const float* g = table + goff;               // saddr + 32-bit voffset
        float*       s = lbuf + (l + (hi ? 1 : 0)) * kD + boff;
        __builtin_amdgcn_global_load_async_to_lds_b128(
            (v4i*)(void*)(const_cast<float*>(g)), (v4i*)(void*)s, 0, 0);
    }
}

// Weighted mean-pool (divided by sum of weights) of L staged rows; weights are wave-uniform (s_load).
template <int L>
__device__ __forceinline__ void consume_rows(const float* __restrict__ w,
                                             const float* __restrict__ lbuf,
                                             int lane, float& ox, float& oy) {
    float ax = 0.f, ay = 0.f, sw = 0.f;
#pragma unroll 10
    for (int l = 0; l < L; ++l) {
        const float wl = w[l];                                     // uniform -> s_load
        const f32x2 e = *(const f32x2*)(lbuf + l * kD + 2 * lane); // ds_load_b64
        ax = fmaf(wl, e.x, ax);
        ay = fmaf(wl, e.y, ay);
        sw += wl;
    }
    const float inv = 1.0f / sw;
    ox = ax * inv;
    oy = ay * inv;
}
#endif  // HAVE_ASYNC_LDS

// Fallback: direct register gather + pool (also near-roofline; ids/weights are
// uniform scalar loads, rows are coalesced global_load_b64 gathers).
template <int L>
__device__ __forceinline__ void pool_direct(const int* __restrict__ ids,
                                            const float* __restrict__ w,
                                            const float* __restrict__ table,
                                            int lane, float& ox, float& oy) {
    float ax = 0.f, ay = 0.f, sw = 0.f;
#pragma unroll 5
    for (int l = 0; l < L; ++l) {
        const int   id = ids[l];   // uniform -> s_load
        const float wl = w[l];     // uniform -> s_load
        const f32x2 e = *(const f32x2*)(table + (size_t)id * kD + 2 * lane);
        ax = fmaf(wl, e.x, ax);
        ay = fmaf(wl, e.y, ay);
        sw += wl;
    }
    const float inv = 1.0f / sw;
    ox = ax * inv;
    oy = ay * inv;
}

__global__ __launch_bounds__(kBlock) void emb_fused_kernel(
    const int* __restrict__ ids_s, const int* __restrict__ ids_a,
    const int* __restrict__ ids_b, const float* __restrict__ w_a,
    const float* __restrict__ w_b, const float* __restrict__ table,
    float* __restrict__ out, int nB) {
#if HAVE_ASYNC_LDS
    __shared__ float lbuf[kWaves][(kLa + kLb) * kD];
#endif
    const int lane = (int)(threadIdx.x & 31);
    const int wave = (int)(threadIdx.x >> 5);
    // One wave per batch row; force the row index into an SGPR so index /
    // weight loads become scalar (KMcnt) and gathers use saddr+voffset form.
    int b = __builtin_amdgcn_readfirstlane((int)(blockIdx.x * kWaves + wave));
    if (b >= nB) return;

    // ---- single-id field: direct gather (table is L2-resident: 25.6MB) ----
    const int   id_s = ids_s[b];  // s_load
    const f32x2 e_s  = *(const f32x2*)(table + (size_t)id_s * kD + 2 * lane);

    float pax, pay, pbx, pby;
#if HAVE_ASYNC_LDS
    float* buf = lbuf[wave];
    // Put all 70 row-gathers in flight on the async DMA path (ASYNCcnt):
    // 25 ops for field a, then 10 for field b. Async loads retire in order,
    // so ASYNCcnt <= 10 proves field a landed -> drain it from LDS while the
    // field-b DMAs are still in flight; then wait 0 and drain field b.
    stage_rows_async<kLa>(ids_a + (size_t)b * kLa, table, buf, lane);
    stage_rows_async<kLb>(ids_b + (size_t)b * kLb, table, buf + kLa * kD, lane);
    wait_async<kLb / 2>();   // field-a rows complete; field-b may be in flight
    consume_rows<kLa>(w_a + (size_t)b * kLa, buf, lane, pax, pay);
    wait_async<0>();         // field-b rows complete
    consume_rows<kLb>(w_b + (size_t)b * kLb, buf + kLa * kD, lane, pbx, pby);
#else
    pool_direct<kLa>(ids_a + (size_t)b * kLa, w_a + (size_t)b * kLa, table,
                     lane, pax, pay);
    pool_direct<kLb>(ids_b + (size_t)b * kLb, w_b + (size_t)b * kLb, table,
                     lane, pbx, pby);
#endif

    // ---- concat [e_single | pool_a | pool_b]; stream out non-temporally so
    // the 12.6MB output doesn't evict the L2-resident embedding table. ----
    float* orow = out + (size_t)b * kOutW;
    const f32x2 v0 = e_s;
    const f32x2 v1 = {pax, pay};
    const f32x2 v2 = {pbx, pby};
    __builtin_nontemporal_store(v0, (f32x2*)(orow + 2 * lane));
    __builtin_nontemporal_store(v1, (f32x2*)(orow + kD + 2 * lane));
    __builtin_nontemporal_store(v2, (f32x2*)(orow + 2 * kD + 2 * lane));
}

extern "C" void kernel_launch(void* const* d_in, const int* in_sizes, int n_in,
                              void* d_out, int out_size, void* d_ws,
                              size_t ws_size, hipStream_t stream) {
    (void)n_in; (void)out_size; (void)d_ws; (void)ws_size;
    const int*   ids_s = (const int*)d_in[0];
    const int*   ids_a = (const int*)d_in[1];
    const int*   ids_b = (const int*)d_in[2];
    const float* w_a   = (const float*)d_in[3];
    const float* w_b   = (const float*)d_in[4];
    const float* table = (const float*)d_in[5];
    float*       out   = (float*)d_out;

    const int nB = in_sizes[0];
    const int blocks = (nB + kWaves - 1) / kWaves;
    emb_fused_kernel<<<blocks, kBlock, 0, stream>>>(ids_s, ids_a, ids_b, w_a,
                                                    w_b, table, out, nB);
}